// TTLinear_88699664597601
// MI455X (gfx1250) — compile-verified
//
#include <hip/hip_runtime.h>

// ---------------------------------------------------------------------------
// Fused tensor-train linear for MI455X (gfx1250, wave32, WMMA).
//
// Reference math (per batch row b, all mode dims = 8, ranks = 1,8,8,8,1):
//   z0[b, o0,r1, i1,i2,i3] = sum_i0 W0[i0][(o0,r1)] * x[b, i0,i1,i2,i3]
//   z1[b,o0, o1,r2, i2,i3] = sum_{r1,i1} W1[(r1,i1)][(o1,r2)] * z0[...]
//   z2[b,o0,o1, o2,r3, i3] = sum_{r2,i2} W2[(r2,i2)][(o2,r3)] * z1[...]
//   y [b, o0,o1,o2,o3]     = sum_{r3,i3} W3[(r3,i3)][o3]      * z2[...]
// Flat core_k storage already matches W_k row-major, no repacking needed.
//
// One workgroup = 4 batch rows, 256 threads = 8 wave32.
// All intermediates live in LDS; fp32 WMMA 16x16x4 for every contraction.
// ---------------------------------------------------------------------------

typedef float v2f __attribute__((ext_vector_type(2)));
typedef float v8f __attribute__((ext_vector_type(8)));

#define WMMA4(A, B, C)                                                         \
  __builtin_amdgcn_wmma_f32_16x16x4_f32(false, (A), false, (B), (short)0,      \
                                        (C), false, false)

#define B_TILE 4

__global__ __launch_bounds__(256, 1) void tt_fused_kernel(
    const float* __restrict__ x, const float* __restrict__ c0,
    const float* __restrict__ c1, const float* __restrict__ c2,
    const float* __restrict__ c3, const float* __restrict__ bias,
    float* __restrict__ out) {
  // Cores in LDS (36 KB)
  __shared__ __align__(16) float lw0[8 * 64];    // [i0][(o0,r1)]
  __shared__ __align__(16) float lw1[64 * 64];   // [(r1,i1)][(o1,r2)]
  __shared__ __align__(16) float lw2[64 * 64];   // [(r2,i2)][(o2,r3)]
  __shared__ __align__(16) float lw3[64 * 8];    // [(r3,i3)][o3]
  // Intermediates (K-contiguous layouts for the *next* stage's A loads)
  __shared__ __align__(16) float z0[2 * 256 * 64];  // [p][(b,s')=256][(r1,i1)]
  __shared__ __align__(16) float z1[8 * 32 * 64];   // [o1][(b,i3)=32][(r2,i2)]
  __shared__ __align__(16) float z2[8 * 32 * 64];   // [o1][(b,o2)=32][(r3,i3)]

  const int tid = threadIdx.x;
  const int w = tid >> 5;          // wave id 0..7
  const int lane = tid & 31;
  const int lo = lane & 15;        // N/M sub-lane
  const int hi = lane >> 4;        // K half-select
  const int bg0 = blockIdx.x * B_TILE;

  // ---- load cores ----
  for (int i = tid; i < 512; i += 256) lw0[i] = c0[i];
  for (int i = tid; i < 4096; i += 256) lw1[i] = c1[i];
  for (int i = tid; i < 4096; i += 256) lw2[i] = c2[i];
  for (int i = tid; i < 512; i += 256) lw3[i] = c3[i];
  __syncthreads();

  for (int pi = 0; pi < 4; ++pi) {  // o0 pair index; o0 = 2*pi + p
    // =====================================================================
    // Stage 0: M=(b,i1,s')=2048, K=i0=8, N=(p,r1)=16.
    //   A[m][i0] = x[bg, i0*512 + i1*64 + s']
    //   B[i0][n] = lw0[i0*64 + pi*16 + n]       (n = p*8 + r1)
    //   D stored as z0[p][(b,s')][(r1,i1)]
    // =====================================================================
    {
      v2f bb0, bb1;
      bb0[0] = lw0[(2 * hi + 0) * 64 + pi * 16 + lo];
      bb0[1] = lw0[(2 * hi + 1) * 64 + pi * 16 + lo];
      bb1[0] = lw0[(4 + 2 * hi + 0) * 64 + pi * 16 + lo];
      bb1[1] = lw0[(4 + 2 * hi + 1) * 64 + pi * 16 + lo];
      for (int mt = w; mt < 128; mt += 8) {
        const int m0 = mt * 16 + lo;  // lane's A row: (b, i1, s')
        const int b = m0 >> 9;
        const int i1 = (m0 >> 6) & 7;
        const int sp = m0 & 63;
        const float* xr = x + (size_t)(bg0 + b) * 4096 + i1 * 64 + sp;
        v2f a0, a1;
        a0[0] = xr[(2 * hi + 0) * 512];
        a0[1] = xr[(2 * hi + 1) * 512];
        a1[0] = xr[(4 + 2 * hi + 0) * 512];
        a1[1] = xr[(4 + 2 * hi + 1) * 512];
        v8f acc = {0.f, 0.f, 0.f, 0.f, 0.f, 0.f, 0.f, 0.f};
        acc = WMMA4(a0, bb0, acc);
        acc = WMMA4(a1, bb1, acc);
        const int p = lo >> 3;
        const int r1 = lo & 7;
#pragma unroll
        for (int v = 0; v < 8; ++v) {
          const int mr = mt * 16 + 8 * hi + v;  // D row: (b, i1, s')
          const int br = mr >> 9;
          const int i1r = (mr >> 6) & 7;
          const int spr = mr & 63;
          z0[p * 16384 + (br * 64 + spr) * 64 + r1 * 8 + i1r] = acc[v];
        }
      }
    }
    __syncthreads();

    for (int p = 0; p < 2; ++p) {
      const int o0 = pi * 2 + p;
      // ===================================================================
      // Stage 1: M=(b,s')=256, K=(r1,i1)=64, N=(o1,r2)=64.
      //   A[m][k] = z0[p][m][k]   (K-contiguous)
      //   B[k][n] = lw1[k*64 + n]
      //   D stored as z1[o1][(b,i3)][(r2,i2)]
      // ===================================================================
      {
        const int nt = w & 3;        // N tile
        const int half = w >> 2;     // M-tile group
        const int nb = nt * 16 + lo;
        v2f wb[16];
#pragma unroll
        for (int t = 0; t < 16; ++t) {
          const int k = 4 * t + 2 * hi;
          wb[t][0] = lw1[k * 64 + nb];
          wb[t][1] = lw1[(k + 1) * 64 + nb];
        }
        const int o1s = nb >> 3;
        const int r2s = nb & 7;
        for (int i = 0; i < 8; ++i) {
          const int mt = half * 8 + i;
          const float* arow = &z0[p * 16384 + (mt * 16 + lo) * 64];
          v8f acc = {0.f, 0.f, 0.f, 0.f, 0.f, 0.f, 0.f, 0.f};
#pragma unroll
          for (int t = 0; t < 16; ++t) {
            v2f a = *(const v2f*)(arow + 4 * t + 2 * hi);
            acc = WMMA4(a, wb[t], acc);
          }
#pragma unroll
          for (int v = 0; v < 8; ++v) {
            const int m1 = mt * 16 + 8 * hi + v;  // (b, s'=(i2,i3))
            const int b = m1 >> 6;
            const int i2 = (m1 >> 3) & 7;
            const int i3 = m1 & 7;
            z1[o1s * 2048 + (b * 8 + i3) * 64 + r2s * 8 + i2] = acc[v];
          }
        }
      }
      __syncthreads();

      // ===================================================================
      // Stage 2: per o1: M=(b,i3)=32, K=(r2,i2)=64, N=(o2,r3)=64.
      //   A[m][k] = z1[o1][m][k]   (K-contiguous)
      //   B[k][n] = lw2[k*64 + n]
      //   D stored as z2[o1][(b,o2)][(r3,i3)]
      // ===================================================================
      {
        const int nt = w & 3;
        const int mt = w >> 2;       // 2 M-tiles across wave groups
        const int nb = nt * 16 + lo;
        v2f wb2[16];
#pragma unroll
        for (int t = 0; t < 16; ++t) {
          const int k = 4 * t + 2 * hi;
          wb2[t][0] = lw2[k * 64 + nb];
          wb2[t][1] = lw2[(k + 1) * 64 + nb];
        }
        const int o2s = nb >> 3;
        const int r3s = nb & 7;
        for (int o1 = 0; o1 < 8; ++o1) {
          const float* arow = &z1[o1 * 2048 + (mt * 16 + lo) * 64];
          v8f acc = {0.f, 0.f, 0.f, 0.f, 0.f, 0.f, 0.f, 0.f};
#pragma unroll
          for (int t = 0; t < 16; ++t) {
            v2f a = *(const v2f*)(arow + 4 * t + 2 * hi);
            acc = WMMA4(a, wb2[t], acc);
          }
#pragma unroll
          for (int v = 0; v < 8; ++v) {
            const int m2 = mt * 16 + 8 * hi + v;  // (b, i3)
            const int b = m2 >> 3;
            const int i3 = m2 & 7;
            z2[o1 * 2048 + (b * 8 + o2s) * 64 + r3s * 8 + i3] = acc[v];
          }
        }
      }
      __syncthreads();

      // ===================================================================
      // Stage 3: per o1 (= wave id): M=(b,o2)=32, K=(r3,i3)=64, N=o3 (8,
      // zero-padded to 16).  D + bias written straight to global memory.
      // ===================================================================
      {
        const int o1 = w;
        v2f wb3[16];
#pragma unroll
        for (int t = 0; t < 16; ++t) {
          const int k = 4 * t + 2 * hi;
          wb3[t][0] = (lo < 8) ? lw3[k * 8 + lo] : 0.f;
          wb3[t][1] = (lo < 8) ? lw3[(k + 1) * 8 + lo] : 0.f;
        }
        for (int mt = 0; mt < 2; ++mt) {
          const float* arow = &z2[o1 * 2048 + (mt * 16 + lo) * 64];
          v8f acc = {0.f, 0.f, 0.f, 0.f, 0.f, 0.f, 0.f, 0.f};
#pragma unroll
          for (int t = 0; t < 16; ++t) {
            v2f a = *(const v2f*)(arow + 4 * t + 2 * hi);
            acc = WMMA4(a, wb3[t], acc);
          }
          if (lo < 8) {
#pragma unroll
            for (int v = 0; v < 8; ++v) {
              const int m3 = mt * 16 + 8 * hi + v;  // (b, o2)
              const int b = m3 >> 3;
              const int o2 = m3 & 7;
              const int feat = ((o0 * 8 + o1) * 8 + o2) * 8 + lo;
              out[(size_t)(bg0 + b) * 4096 + feat] = acc[v] + bias[feat];
            }
          }
        }
      }
      __syncthreads();
    }
  }
}

extern "C" void kernel_launch(void* const* d_in, const int* in_sizes, int n_in,
                              void* d_out, int out_size, void* d_ws,
                              size_t ws_size, hipStream_t stream) {
  const float* x = (const float*)d_in[0];
  const float* c0 = (const float*)d_in[1];
  const float* c1 = (const float*)d_in[2];
  const float* c2 = (const float*)d_in[3];
  const float* c3 = (const float*)d_in[4];
  const float* bias = (const float*)d_in[5];
  float* out = (float*)d_out;

  dim3 grid(4096 / B_TILE);  // 1024 workgroups, 4 batch rows each
  dim3 block(256);           // 8 wave32 per workgroup
  tt_fused_kernel<<<grid, block, 0, stream>>>(x, c0, c1, c2, c3, bias, out);
}